// GNN_68839735820556
// MI455X (gfx1250) — compile-verified
//
#include <hip/hip_runtime.h>
#include <hip/hip_bf16.h>

typedef __attribute__((ext_vector_type(2))) float v2f;
typedef __attribute__((ext_vector_type(8))) float v8f;

// ---------------------------------------------------------------- utilities
__global__ void k_zero_i32(int* __restrict__ p, int n) {
    int i = blockIdx.x * blockDim.x + threadIdx.x;
    if (i < n) p[i] = 0;
}

// count in-degree per dst node
__global__ void k_count(const int* __restrict__ dst, int* __restrict__ cnt, int E) {
    int e = blockIdx.x * blockDim.x + threadIdx.x;
    if (e < E) atomicAdd(&cnt[dst[e]], 1);
}

// single-block exclusive scan of cnt[0..n) -> rowptr[0..n]
__global__ void k_scan(const int* __restrict__ cnt, int* __restrict__ rowptr, int n) {
    __shared__ int sm[1024];
    __shared__ int run;
    int tid = threadIdx.x;
    if (tid == 0) run = 0;
    __syncthreads();
    for (int base = 0; base < n; base += 1024) {
        int i = base + tid;
        int v = (i < n) ? cnt[i] : 0;
        sm[tid] = v;
        __syncthreads();
        for (int off = 1; off < 1024; off <<= 1) {
            int t = (tid >= off) ? sm[tid - off] : 0;
            __syncthreads();
            sm[tid] += t;
            __syncthreads();
        }
        int excl = sm[tid] - v;
        int r = run;
        if (i < n) rowptr[i] = r + excl;
        __syncthreads();
        if (tid == 1023) run = r + sm[1023];
        __syncthreads();
    }
    if (tid == 0) rowptr[n] = run;
}

// scatter edge ids into CSR slots
__global__ void k_fill(const int* __restrict__ dst, const int* __restrict__ rowptr,
                       int* __restrict__ cursor, int* __restrict__ eidx, int E) {
    int e = blockIdx.x * blockDim.x + threadIdx.x;
    if (e >= E) return;
    int d = dst[e];
    int pos = atomicAdd(&cursor[d], 1);
    eidx[rowptr[d] + pos] = e;
}

// per-node insertion sort of edge ids -> deterministic summation order
__global__ void k_sort(int* __restrict__ eidx, const int* __restrict__ rowptr, int n) {
    int i = blockIdx.x * blockDim.x + threadIdx.x;
    if (i >= n) return;
    int beg = rowptr[i], end = rowptr[i + 1];
    for (int a = beg + 1; a < end; ++a) {
        int key = eidx[a];
        int b = a - 1;
        while (b >= beg && eidx[b] > key) { eidx[b + 1] = eidx[b]; --b; }
        eidx[b + 1] = key;
    }
}

// deg (incl. self-loop weight 1) and dis = rsqrt(deg)
__global__ void k_deg(const float* __restrict__ ew, const int* __restrict__ rowptr,
                      const int* __restrict__ eidx, float* __restrict__ dis, int n) {
    int i = blockIdx.x * blockDim.x + threadIdx.x;
    if (i >= n) return;
    float d = 1.0f;                       // self-loop weight
    int beg = rowptr[i], end = rowptr[i + 1];
    for (int k = beg; k < end; ++k) d += ew[eidx[k]];
    dis[i] = (d > 0.f) ? rsqrtf(d) : 0.f;
}

// per-CSR-entry: source node id and symmetric-normalized weight
__global__ void k_norm(const int* __restrict__ src, const float* __restrict__ ew,
                       const int* __restrict__ rowptr, const int* __restrict__ eidx,
                       const float* __restrict__ dis, int* __restrict__ csrc,
                       float* __restrict__ cnorm, int n) {
    int i = blockIdx.x * blockDim.x + threadIdx.x;
    if (i >= n) return;
    float di = dis[i];
    int beg = rowptr[i], end = rowptr[i + 1];
    for (int k = beg; k < end; ++k) {
        int e = eidx[k];
        int s = src[e];
        csrc[k]  = s;
        cnorm[k] = dis[s] * ew[e] * di;
    }
}

// ---------------------------------------------------------------- WMMA GEMM
// Hout[npad x 64] = A[n x 64] @ W[64 x 64], fp32 WMMA.
// One wave per 16-row stripe: A fragment (32 VGPRs) loaded once, branch-free
// (clamped row + 0/1 mask); reused across 4 column tiles; 16 back-to-back
// WMMAs per tile. Hout is padded to a 16-row multiple, so stores are
// unconditional (no EXEC save/restore in the epilogue).
__global__ void k_gemm64(const float* __restrict__ A, const float* __restrict__ W,
                         float* __restrict__ Hout, int n) {
    int wave = (int)((blockIdx.x * blockDim.x + threadIdx.x) >> 5);
    int lane = threadIdx.x & 31;
    int tiles_m = (n + 15) >> 4;
    if (wave >= tiles_m) return;
    int hl = lane >> 4;                   // half: 0 -> K{4k,4k+1}, 1 -> K{4k+2,4k+3}
    int l  = lane & 15;
    int arow  = wave * 16 + l;
    float amask = (arow < n) ? 1.0f : 0.0f;   // branch-free ragged-edge handling
    int   arows = (arow < n) ? arow : 0;      // clamp: loads never OOB
    const float* ap = A + (size_t)arows * 64 + hl * 2;

    v2f afrag[16];
#pragma unroll
    for (int kk = 0; kk < 16; ++kk) {     // 16 clustered b64 loads, no EXEC games
        afrag[kk].x = ap[kk * 4]     * amask;
        afrag[kk].y = ap[kk * 4 + 1] * amask;
    }

#pragma unroll 1
    for (int tn = 0; tn < 4; ++tn) {
        const float* wp = W + (hl * 2) * 64 + tn * 16 + l;
        v2f bfrag[16];
#pragma unroll
        for (int kk = 0; kk < 16; ++kk) {
            bfrag[kk].x = wp[(kk * 4) * 64];
            bfrag[kk].y = wp[(kk * 4 + 1) * 64];
        }
        v8f c = {};
#pragma unroll
        for (int kk = 0; kk < 16; ++kk) {
            // (neg_a, A, neg_b, B, c_mod, C, reuse_a, reuse_b)
            c = __builtin_amdgcn_wmma_f32_16x16x4_f32(false, afrag[kk], false, bfrag[kk],
                                                      (short)0, c, false, false);
        }
        // C layout: lane<16 -> rows rbase..rbase+7 (cols l), lane>=16 -> rows +8
        float* op = Hout + (size_t)(wave * 16 + hl * 8) * 64 + tn * 16 + l;
#pragma unroll
        for (int m = 0; m < 8; ++m) op[m * 64] = c[m];   // unconditional (padded dst)
    }
}

// -------------------------------------------------- fused aggregate+bias+relu
// one wave32 per node; lane owns features {lane, lane+32}; no atomics.
__global__ void k_aggregate(const float* __restrict__ h, const int* __restrict__ rowptr,
                            const int* __restrict__ csrc, const float* __restrict__ cnorm,
                            const float* __restrict__ dis, const float* __restrict__ bias,
                            float* __restrict__ out, int n, int relu) {
    int node = (int)((blockIdx.x * blockDim.x + threadIdx.x) >> 5);
    int lane = threadIdx.x & 31;
    if (node >= n) return;
    float ds = dis[node]; ds *= ds;       // self-loop norm: dis*1*dis
    const float* hp = h + (size_t)node * 64;
    float a0 = ds * hp[lane];
    float a1 = ds * hp[lane + 32];
    int beg = rowptr[node], end = rowptr[node + 1];
    for (int k = beg; k < end; ++k) {
        int s = csrc[k];
        float w = cnorm[k];
        const float* hs = h + (size_t)s * 64;
        int kn = (k + 1 < end) ? k + 1 : k;   // branch-free next-row prefetch
        __builtin_prefetch(h + (size_t)csrc[kn] * 64 + lane, 0, 3);
        a0 += w * hs[lane];
        a1 += w * hs[lane + 32];
    }
    a0 += bias[lane];
    a1 += bias[lane + 32];
    if (relu) { a0 = fmaxf(a0, 0.f); a1 = fmaxf(a1, 0.f); }
    out[(size_t)node * 64 + lane]      = a0;
    out[(size_t)node * 64 + lane + 32] = a1;
}

// ---------------------------------------------- pooling (batch ids sorted)
__global__ void k_pool(const float* __restrict__ agg, const int* __restrict__ batch,
                       float* __restrict__ pooled, float* __restrict__ pcnt, int n) {
    int g = blockIdx.x;
    __shared__ int slo, shi;
    if (threadIdx.x == 0) {
        int lo = 0, hi = n;
        while (lo < hi) { int mid = (lo + hi) >> 1; if (batch[mid] < g) lo = mid + 1; else hi = mid; }
        slo = lo;
        hi = n;
        while (lo < hi) { int mid = (lo + hi) >> 1; if (batch[mid] < g + 1) lo = mid + 1; else hi = mid; }
        shi = lo;
    }
    __syncthreads();
    int lo = slo, hi = shi;
    int f = threadIdx.x;                  // 64 threads = 64 features
    float s = 0.f;
    for (int i = lo; i < hi; ++i) s += agg[(size_t)i * 64 + f];
    pooled[g * 64 + f] = s;
    if (f == 0) pcnt[g] = (float)(hi - lo);
}

__global__ void k_final(const float* __restrict__ pooled, const float* __restrict__ pcnt,
                        const float* __restrict__ Wl, const float* __restrict__ bl,
                        float* __restrict__ out, int G, int C) {
    int t = threadIdx.x;
    if (t >= G * C) return;
    int g = t / C, c = t % C;
    float s = 0.f;
    for (int f = 0; f < 64; ++f) s += pooled[g * 64 + f] * Wl[f * C + c];
    out[t] = s / fmaxf(pcnt[g], 1.f) + bl[c];
}

// ---------------------------------------------------------------- launcher
extern "C" void kernel_launch(void* const* d_in, const int* in_sizes, int n_in,
                              void* d_out, int out_size, void* d_ws, size_t ws_size,
                              hipStream_t stream) {
    const float* x     = (const float*)d_in[0];
    const int*   eix   = (const int*)d_in[1];     // (2,E): [src | dst]
    const float* ew    = (const float*)d_in[2];
    const int*   batch = (const int*)d_in[3];
    const float* W1 = (const float*)d_in[4];  const float* b1 = (const float*)d_in[5];
    const float* W2 = (const float*)d_in[6];  const float* b2 = (const float*)d_in[7];
    const float* W3 = (const float*)d_in[8];  const float* b3 = (const float*)d_in[9];
    const float* Wl = (const float*)d_in[10]; const float* bl = (const float*)d_in[11];
    float* out = (float*)d_out;

    const int N = in_sizes[0] / 64;               // nodes (F = 64)
    const int E = in_sizes[2];                    // edges
    const int C = 2;
    const int G = out_size / C;                   // graphs
    const int Npad = ((N + 15) / 16) * 16;        // 16-row padded for guard-free stores
    const int* src = eix;
    const int* dst = eix + E;

    // workspace carve-up (256B aligned)
    char* p = (char*)d_ws;
    size_t off = 0;
    auto carve = [&](size_t bytes) { char* r = p + off; off = (off + bytes + 255) & ~(size_t)255; return r; };
    int*   cnt    = (int*)  carve((size_t)2 * N * sizeof(int));   // cnt + cursor adjacent
    int*   cursor = cnt + N;
    int*   rowptr = (int*)  carve((size_t)(N + 1) * sizeof(int));
    int*   eidx   = (int*)  carve((size_t)E * sizeof(int));
    int*   csrc   = (int*)  carve((size_t)E * sizeof(int));
    float* cnorm  = (float*)carve((size_t)E * sizeof(float));
    float* dis    = (float*)carve((size_t)N * sizeof(float));
    float* hbuf   = (float*)carve((size_t)Npad * 64 * sizeof(float));
    float* agg    = (float*)carve((size_t)Npad * 64 * sizeof(float));
    float* pooled = (float*)carve((size_t)G * 64 * sizeof(float));
    float* pcnt   = (float*)carve((size_t)G * sizeof(float));
    (void)ws_size; (void)n_in;

    const int T = 256;
    // ---- graph preprocessing (deterministic CSR by dst)
    k_zero_i32<<<(2 * N + T - 1) / T, T, 0, stream>>>(cnt, 2 * N);
    k_count   <<<(E + T - 1) / T, T, 0, stream>>>(dst, cnt, E);
    k_scan    <<<1, 1024, 0, stream>>>(cnt, rowptr, N);
    k_fill    <<<(E + T - 1) / T, T, 0, stream>>>(dst, rowptr, cursor, eidx, E);
    k_sort    <<<(N + T - 1) / T, T, 0, stream>>>(eidx, rowptr, N);
    k_deg     <<<(N + T - 1) / T, T, 0, stream>>>(ew, rowptr, eidx, dis, N);
    k_norm    <<<(N + T - 1) / T, T, 0, stream>>>(src, ew, rowptr, eidx, dis, csrc, cnorm, N);

    // ---- 3 GCN layers: WMMA GEMM + fused gather/bias/relu
    int tiles_m = Npad / 16;                      // one wave per 16-row stripe
    int gblocks = (tiles_m * 32 + T - 1) / T;
    int ablocks = (N * 32 + T - 1) / T;           // one wave per node

    k_gemm64   <<<gblocks, T, 0, stream>>>(x,   W1, hbuf, N);
    k_aggregate<<<ablocks, T, 0, stream>>>(hbuf, rowptr, csrc, cnorm, dis, b1, agg, N, 1);
    k_gemm64   <<<gblocks, T, 0, stream>>>(agg, W2, hbuf, N);
    k_aggregate<<<ablocks, T, 0, stream>>>(hbuf, rowptr, csrc, cnorm, dis, b2, agg, N, 1);
    k_gemm64   <<<gblocks, T, 0, stream>>>(agg, W3, hbuf, N);
    k_aggregate<<<ablocks, T, 0, stream>>>(hbuf, rowptr, csrc, cnorm, dis, b3, agg, N, 0);

    // ---- mean pool (sorted batch, no atomics) + linear head
    k_pool <<<G, 64, 0, stream>>>(agg, batch, pooled, pcnt, N);
    k_final<<<1, 64, 0, stream>>>(pooled, pcnt, Wl, bl, out, G, C);
}